// HETGraphConv_30657476559616
// MI455X (gfx1250) — compile-verified
//
#include <hip/hip_runtime.h>

#define ND 100000
#define NT 100000
#define NE 500000
#define DD 128

typedef __attribute__((ext_vector_type(2))) float v2f;
typedef __attribute__((ext_vector_type(8))) float v8f;

__global__ __launch_bounds__(256) void zero_f32(float* __restrict__ p, long n) {
  long i = (long)blockIdx.x * blockDim.x + threadIdx.x;
  long stride = (long)gridDim.x * blockDim.x;
  for (; i < n; i += stride) p[i] = 0.0f;
}

__global__ __launch_bounds__(256) void degree_kernel(const int* __restrict__ src,
                                                     const int* __restrict__ dst,
                                                     float* __restrict__ degO,
                                                     float* __restrict__ degI, int e) {
  int i = blockIdx.x * blockDim.x + threadIdx.x;
  if (i < e) {
    atomicAdd(&degO[src[i]], 1.0f);
    atomicAdd(&degI[dst[i]], 1.0f);
  }
}

// Y[M,128] = (diag(rsqrt(max(degO,1))) * X[M,128]) @ W[128,128]   (fp32 WMMA 16x16x4)
// Block: 128 threads = 4 waves; block covers 64 rows; wave w -> rows [16w,16w+16).
__global__ __launch_bounds__(128) void gemm_rowscale_wmma(const float* __restrict__ X,
                                                          const float* __restrict__ W,
                                                          const float* __restrict__ degO,
                                                          float* __restrict__ Y, int M) {
  __shared__ float Alds[64][129];   // padded: conflict-free column reads
  __shared__ float Wlds[128][16];

  const int tid  = threadIdx.x;
  const int w    = tid >> 5;        // wave id 0..3
  const int l    = tid & 31;        // lane id
  const int half = l >> 4;          // 0: lanes 0-15, 1: lanes 16-31
  const int mn   = l & 15;          // row/col within 16
  const int rBase = blockIdx.x * 64;

  // Stage A with out-degree row scaling (rows >= M -> 0)
  for (int idx = tid; idx < 64 * 128; idx += 128) {
    int r = idx >> 7, c = idx & 127;
    int row = rBase + r;
    float v = 0.0f;
    if (row < M) v = X[(size_t)row * DD + c] * rsqrtf(fmaxf(degO[row], 1.0f));
    Alds[r][c] = v;
  }
  __syncthreads();

  const int rw = 16 * w + mn;       // A-matrix row this lane supplies

  for (int nt = 0; nt < 8; ++nt) {  // 8 N-tiles of 16 columns
    const int c0 = nt * 16;
    for (int idx = tid; idx < 128 * 16; idx += 128) {
      int r = idx >> 4, c = idx & 15;
      Wlds[r][c] = W[(size_t)r * DD + c0 + c];
    }
    __syncthreads();

    v8f acc = {0.f, 0.f, 0.f, 0.f, 0.f, 0.f, 0.f, 0.f};
    #pragma unroll
    for (int kk = 0; kk < 128; kk += 4) {
      // A 16x4 frag: lanes 0-15 -> K=kk,kk+1 ; lanes 16-31 -> K=kk+2,kk+3
      // B 4x16 frag: lanes 0-15 -> rows kk,kk+1 ; lanes 16-31 -> rows kk+2,kk+3
      const int k0 = kk + 2 * half;
      v2f a, b;
      a.x = Alds[rw][k0];
      a.y = Alds[rw][k0 + 1];
      b.x = Wlds[k0][mn];
      b.y = Wlds[k0 + 1][mn];
      acc = __builtin_amdgcn_wmma_f32_16x16x4_f32(false, a, false, b,
                                                  (short)0, acc, false, false);
    }

    // C 16x16: VGPR v -> row (v + 8*half), col mn
    #pragma unroll
    for (int v = 0; v < 8; ++v) {
      int row = rBase + 16 * w + v + 8 * half;
      if (row < M) Y[(size_t)row * DD + c0 + mn] = acc[v];
    }
    __syncthreads();
  }
}

// One wave per edge: H[dst] += Y[src] * rsqrt(max(degI[dst],1)); lane handles 4 floats.
__global__ __launch_bounds__(256) void scatter_kernel(const float* __restrict__ Y,
                                                      const int* __restrict__ src,
                                                      const int* __restrict__ dst,
                                                      const float* __restrict__ degI,
                                                      float* __restrict__ H, int e) {
  int wave = blockIdx.x * 8 + (threadIdx.x >> 5);
  int lane = threadIdx.x & 31;
  if (wave < e) {
    int s = src[wave], d = dst[wave];
    float sc = rsqrtf(fmaxf(degI[d], 1.0f));
    float4 y = ((const float4*)(Y + (size_t)s * DD))[lane];
    float* h = H + (size_t)d * DD + lane * 4;
    atomicAdd(h + 0, y.x * sc);
    atomicAdd(h + 1, y.y * sc);
    atomicAdd(h + 2, y.z * sc);
    atomicAdd(h + 3, y.w * sc);
  }
}

// Column-parallel partial sums: thread = column, blocks stride rows (coalesced).
__global__ __launch_bounds__(128) void bn_stats(const float* __restrict__ H, int M,
                                                float* __restrict__ sum,
                                                float* __restrict__ sq) {
  int c = threadIdx.x;
  float s = 0.0f, q = 0.0f;
  for (int r = blockIdx.x; r < M; r += gridDim.x) {
    float v = H[(size_t)r * DD + c];
    s += v;
    q += v * v;
  }
  atomicAdd(&sum[c], s);
  atomicAdd(&sq[c], q);
}

__global__ __launch_bounds__(256) void bn_apply(float* __restrict__ H, int M,
                                                const float* __restrict__ sum,
                                                const float* __restrict__ sq,
                                                const float* __restrict__ gamma,
                                                const float* __restrict__ beta) {
  long n = (long)M * DD;
  long i = (long)blockIdx.x * blockDim.x + threadIdx.x;
  long stride = (long)gridDim.x * blockDim.x;
  float invM = 1.0f / (float)M;
  for (; i < n; i += stride) {
    int c = (int)(i & (DD - 1));
    float mean = sum[c] * invM;
    float var  = sq[c] * invM - mean * mean;
    H[i] = (H[i] - mean) * rsqrtf(var + 1e-5f) * gamma[c] + beta[c];
  }
}

extern "C" void kernel_launch(void* const* d_in, const int* in_sizes, int n_in,
                              void* d_out, int out_size, void* d_ws, size_t ws_size,
                              hipStream_t stream) {
  const float* x_d  = (const float*)d_in[0];
  const float* x_t  = (const float*)d_in[1];
  const float* W_dd = (const float*)d_in[2];
  const float* W_tt = (const float*)d_in[4];
  const float* W_dt = (const float*)d_in[6];
  const float* gamma = (const float*)d_in[8];
  const float* beta  = (const float*)d_in[9];
  // biases b_dd/b_tt/b_dt (d_in[3],[5],[7]) provably cancel under batch-stats BN.
  const int* src_dd = (const int*)d_in[10];
  const int* dst_dd = (const int*)d_in[11];
  const int* src_tt = (const int*)d_in[12];
  const int* dst_tt = (const int*)d_in[13];
  const int* src_dt = (const int*)d_in[14];
  const int* dst_dt = (const int*)d_in[15];
  const int* src_td = (const int*)d_in[16];
  const int* dst_td = (const int*)d_in[17];

  float* out = (float*)d_out;
  float* h_d = out;
  float* h_t = out + (size_t)ND * DD;

  float* ws = (float*)d_ws;
  float* Y   = ws;                      // 12.8M floats (reused per relation)
  float* deg = ws + (size_t)ND * DD;    // 8 x 100000 degree arrays
  float* degO_dd = deg;
  float* degI_dd = deg + 100000;
  float* degO_tt = deg + 200000;
  float* degI_tt = deg + 300000;
  float* degO_dt = deg + 400000;
  float* degI_dt = deg + 500000;
  float* degO_td = deg + 600000;
  float* degI_td = deg + 700000;
  float* stats = deg + 800000;          // 512 floats
  float* sum_d = stats;
  float* sq_d  = stats + 128;
  float* sum_t = stats + 256;
  float* sq_t  = stats + 384;

  // Zero accumulators every call (output is an atomic accumulation target).
  zero_f32<<<1024, 256, 0, stream>>>(deg, 800512L);
  zero_f32<<<4096, 256, 0, stream>>>(out, (long)(ND + NT) * DD);

  const int degBlocks = (NE + 255) / 256;
  degree_kernel<<<degBlocks, 256, 0, stream>>>(src_dd, dst_dd, degO_dd, degI_dd, NE);
  degree_kernel<<<degBlocks, 256, 0, stream>>>(src_tt, dst_tt, degO_tt, degI_tt, NE);
  degree_kernel<<<degBlocks, 256, 0, stream>>>(src_dt, dst_dt, degO_dt, degI_dt, NE);
  degree_kernel<<<degBlocks, 256, 0, stream>>>(src_td, dst_td, degO_td, degI_td, NE);

  const int gemmBlocksD = (ND + 63) / 64;
  const int gemmBlocksT = (NT + 63) / 64;
  const int scatBlocks  = (NE + 7) / 8;

  // dd: x_d --W_dd--> h_d
  gemm_rowscale_wmma<<<gemmBlocksD, 128, 0, stream>>>(x_d, W_dd, degO_dd, Y, ND);
  scatter_kernel<<<scatBlocks, 256, 0, stream>>>(Y, src_dd, dst_dd, degI_dd, h_d, NE);
  // td: x_t --W_dt--> h_d
  gemm_rowscale_wmma<<<gemmBlocksT, 128, 0, stream>>>(x_t, W_dt, degO_td, Y, NT);
  scatter_kernel<<<scatBlocks, 256, 0, stream>>>(Y, src_td, dst_td, degI_td, h_d, NE);
  // tt: x_t --W_tt--> h_t
  gemm_rowscale_wmma<<<gemmBlocksT, 128, 0, stream>>>(x_t, W_tt, degO_tt, Y, NT);
  scatter_kernel<<<scatBlocks, 256, 0, stream>>>(Y, src_tt, dst_tt, degI_tt, h_t, NE);
  // dt: x_d --W_dt--> h_t
  gemm_rowscale_wmma<<<gemmBlocksD, 128, 0, stream>>>(x_d, W_dt, degO_dt, Y, ND);
  scatter_kernel<<<scatBlocks, 256, 0, stream>>>(Y, src_dt, dst_dt, degI_dt, h_t, NE);

  // Shared BatchNorm1d (training-mode batch stats), applied per type.
  bn_stats<<<512, 128, 0, stream>>>(h_d, ND, sum_d, sq_d);
  bn_stats<<<512, 128, 0, stream>>>(h_t, NT, sum_t, sq_t);
  bn_apply<<<4096, 256, 0, stream>>>(h_d, ND, sum_d, sq_d, gamma, beta);
  bn_apply<<<4096, 256, 0, stream>>>(h_t, NT, sum_t, sq_t, gamma, beta);
}